// GCNEncoder_19997367730401
// MI455X (gfx1250) — compile-verified
//
#include <hip/hip_runtime.h>

// ---------------------------------------------------------------------------
// 2-layer GCN forward for MI455X (gfx1250).
//  - fp32 WMMA (v_wmma_f32_16x16x4_f32) for both dense GEMMs (exact fp32 math)
//    with the B slab register-cached and reused across 4 M-tiles per wave.
//  - wave-per-edge b128 gather + hardware global_atomic_add_f32 scatter; all
//    intermediates (<= 51MB each) stay resident in the 192MB L2.
// ---------------------------------------------------------------------------

typedef float v2f __attribute__((ext_vector_type(2)));
typedef float v4f __attribute__((ext_vector_type(4)));
typedef float v8f __attribute__((ext_vector_type(8)));

__device__ __forceinline__ void atomicAddF32(float* p, float v) {
  unsafeAtomicAdd(p, v);   // lowers to global_atomic_add_f32 on gfx1250
}

// ---- degree / normalization ------------------------------------------------
__global__ void k_init_deg(float* __restrict__ deg, int N) {
  int i = blockIdx.x * blockDim.x + threadIdx.x;
  if (i < N) deg[i] = 1.0f;                       // self loop contributes 1
}

__global__ void k_deg_edges(float* __restrict__ deg, const int* __restrict__ dst, int E) {
  int i = blockIdx.x * blockDim.x + threadIdx.x;
  if (i < E) atomicAddF32(&deg[dst[i]], 1.0f);
}

__global__ void k_rsqrt(float* __restrict__ deg, int N) {
  int i = blockIdx.x * blockDim.x + threadIdx.x;
  if (i < N) deg[i] = rsqrtf(deg[i]);             // deg >= 1 always
}

// ---- fp32 WMMA GEMM: Y[M,Nout] = X[M,K] @ W[K,Nout] ------------------------
// One wave owns one 16-col N-tile and MT consecutive 16-row M-tiles.
// The K x 16 B slab is register-cached once (K/2 VGPRs) and reused MT times.
// VGPR layouts per CDNA5 ISA 7.12.2:
//   A (16x4):  lanes 0-15: row=lane, K={k0,k0+1}; lanes 16-31: K={k0+2,k0+3}
//   B (4x16):  lanes 0-15: col=lane, K={k0,k0+1}; lanes 16-31: K={k0+2,k0+3}
//   C/D:       VGPR v, lane(half,r): element (row = v + 8*half, col = r)
template <int K, int MT>
__global__ void k_gemm_wmma(const float* __restrict__ X, const float* __restrict__ W,
                            float* __restrict__ Y, int M, int Nout) {
  constexpr int KSTEPS = K / 4;
  const int ntiles  = Nout >> 4;
  const int mtiles  = (M + 15) >> 4;
  const int mgroups = (mtiles + MT - 1) / MT;
  const int wave    = blockIdx.x * (blockDim.x >> 5) + (threadIdx.x >> 5);
  if (wave >= mgroups * ntiles) return;           // uniform per wave: EXEC stays full
  const int mg   = wave / ntiles;
  const int nt   = wave - mg * ntiles;
  const int lane = threadIdx.x & 31;
  const int half = lane >> 4;
  const int r    = lane & 15;
  const int n0   = nt << 4;

  // ---- preload B slab for this N-tile (reused across MT M-tiles) ----------
  const float* Bptr = W + (size_t)(2 * half) * Nout + (n0 + r);
  v2f bq[KSTEPS];
#pragma unroll
  for (int kk = 0; kk < KSTEPS; ++kk) {
    bq[kk][0] = Bptr[(size_t)(4 * kk + 0) * Nout];
    bq[kk][1] = Bptr[(size_t)(4 * kk + 1) * Nout];
  }

  for (int t = 0; t < MT; ++t) {
    const int mt = mg * MT + t;
    if (mt >= mtiles) break;                      // uniform across wave
    const int m0 = mt << 4;
    int arow = m0 + r;
    if (arow >= M) arow = M - 1;                  // clamp; rows discarded at store
    const float* Aptr = X + (size_t)arow * K + 2 * half;

    v8f c = {};
#pragma unroll
    for (int kk = 0; kk < KSTEPS; ++kk) {
      v2f a = *(const v2f*)(Aptr + 4 * kk);       // global_load_b64, K-contiguous
      c = __builtin_amdgcn_wmma_f32_16x16x4_f32(
          /*neg_a=*/false, a, /*neg_b=*/false, bq[kk],
          /*c_mod=*/(short)0, c, /*reuse_a=*/false, /*reuse_b=*/false);
    }
#pragma unroll
    for (int v = 0; v < 8; ++v) {
      int row = m0 + v + 8 * half;
      if (row < M) Y[(size_t)row * Nout + n0 + r] = c[v];
    }
  }
}

// ---- self-loop init: Y = X * dinv^2  (writes every element of Y) -----------
__global__ void k_self_loop(float* __restrict__ Y, const float* __restrict__ X,
                            const float* __restrict__ dinv, long long total, int logD) {
  long long t = (long long)blockIdx.x * blockDim.x + threadIdx.x;
  if (t < total) {
    float w = dinv[t >> logD];
    Y[t] = X[t] * w * w;
  }
}

// ---- wave-per-edge normalized scatter-add (b128 gather, b32 fp atomics) ----
__global__ void k_scatter(float* __restrict__ Y, const float* __restrict__ X,
                          const float* __restrict__ dinv,
                          const int* __restrict__ src, const int* __restrict__ dst,
                          int E, int D) {
  const int e = blockIdx.x * (blockDim.x >> 5) + (threadIdx.x >> 5);
  if (e >= E) return;
  const int lane = threadIdx.x & 31;
  const int s = src[e], d = dst[e];               // same addr across wave -> broadcast
  const float nrm = dinv[s] * dinv[d];
  const float* xs = X + (size_t)s * D;
  float* yd = Y + (size_t)d * D;
  if (D == 128) {                                 // lane covers 4 contiguous floats
    v4f x = ((const v4f*)xs)[lane];               // global_load_b128
    float* p = yd + 4 * lane;
    atomicAddF32(p + 0, x[0] * nrm);
    atomicAddF32(p + 1, x[1] * nrm);
    atomicAddF32(p + 2, x[2] * nrm);
    atomicAddF32(p + 3, x[3] * nrm);
  } else if (D == 64) {                           // lane covers 2 contiguous floats
    v2f x = ((const v2f*)xs)[lane];               // global_load_b64
    float* p = yd + 2 * lane;
    atomicAddF32(p + 0, x[0] * nrm);
    atomicAddF32(p + 1, x[1] * nrm);
  } else {                                        // generic fallback
    for (int j = lane; j < D; j += 32)
      atomicAddF32(&yd[j], xs[j] * nrm);
  }
}

// ---- bias (+ optional relu), in place --------------------------------------
__global__ void k_bias_relu(float* __restrict__ Y, const float* __restrict__ b,
                            long long total, int Dmask) {
  long long t = (long long)blockIdx.x * blockDim.x + threadIdx.x;
  if (t < total) {
    float v = Y[t] + b[t & Dmask];
    Y[t] = fmaxf(v, 0.0f);
  }
}

__global__ void k_bias_add(float* __restrict__ Y, const float* __restrict__ b,
                           long long total, int Dmask) {
  long long t = (long long)blockIdx.x * blockDim.x + threadIdx.x;
  if (t < total) Y[t] = Y[t] + b[t & Dmask];
}

// ---------------------------------------------------------------------------
extern "C" void kernel_launch(void* const* d_in, const int* in_sizes, int n_in,
                              void* d_out, int out_size, void* d_ws, size_t ws_size,
                              hipStream_t stream) {
  const float* emb  = (const float*)d_in[0];
  const float* W1   = (const float*)d_in[1];
  const float* b1   = (const float*)d_in[2];
  const float* W2   = (const float*)d_in[3];
  const float* b2   = (const float*)d_in[4];
  const int*   edge = (const int*)d_in[5];

  const int D_OUT = in_sizes[4];            // 64  (b2 length = D_EMB)
  const int D_HID = in_sizes[2];            // 128 (b1 length)
  const int N     = in_sizes[0] / D_OUT;    // 100000
  const int E     = in_sizes[5] / 2;        // 1.6M
  const int* src  = edge;                   // edge_index row 0
  const int* dst  = edge + E;               // edge_index row 1

  // workspace layout: dinv | x (reused x1 then x2) | agg
  char* ws = (char*)d_ws;
  const size_t o1 = (((size_t)N * 4) + 255) & ~(size_t)255;
  const size_t o2 = o1 + ((((size_t)N * D_HID * 4) + 255) & ~(size_t)255);
  float* dinv = (float*)ws;
  float* xbuf = (float*)(ws + o1);          // x1 (N*128), later x2 (N*64)
  float* agg  = (float*)(ws + o2);          // layer-1 aggregation / hidden h

  const int T = 256;                        // 8 waves per block (wave32)
  const int wavesPerBlk = T / 32;
  const int mtiles  = (N + 15) / 16;
  const int mgroups = (mtiles + 3) / 4;     // MT = 4
  const long long nh = (long long)N * D_HID;
  const long long no = (long long)N * D_OUT;

  // --- normalization --------------------------------------------------------
  k_init_deg <<<(N + T - 1) / T, T, 0, stream>>>(dinv, N);
  k_deg_edges<<<(E + T - 1) / T, T, 0, stream>>>(dinv, dst, E);
  k_rsqrt    <<<(N + T - 1) / T, T, 0, stream>>>(dinv, N);

  // --- layer 1: x1 = emb @ W1 ; agg = scatter(norm * x1[src]) ; h = relu(agg+b1)
  {
    int waves = mgroups * (D_HID / 16);
    k_gemm_wmma<64, 4><<<(waves + wavesPerBlk - 1) / wavesPerBlk, T, 0, stream>>>(
        emb, W1, xbuf, N, D_HID);
    k_self_loop<<<(int)((nh + T - 1) / T), T, 0, stream>>>(agg, xbuf, dinv, nh, 7);
    k_scatter  <<<(E + wavesPerBlk - 1) / wavesPerBlk, T, 0, stream>>>(
        agg, xbuf, dinv, src, dst, E, D_HID);
    k_bias_relu<<<(int)((nh + T - 1) / T), T, 0, stream>>>(agg, b1, nh, D_HID - 1);
  }

  // --- layer 2: x2 = h @ W2 ; out = scatter(norm * x2[src]) + b2 ------------
  {
    int waves = mgroups * (D_OUT / 16);
    k_gemm_wmma<128, 4><<<(waves + wavesPerBlk - 1) / wavesPerBlk, T, 0, stream>>>(
        agg, W2, xbuf, N, D_OUT);
    float* out = (float*)d_out;
    k_self_loop<<<(int)((no + T - 1) / T), T, 0, stream>>>(out, xbuf, dinv, no, 6);
    k_scatter  <<<(E + wavesPerBlk - 1) / wavesPerBlk, T, 0, stream>>>(
        out, xbuf, dinv, src, dst, E, D_OUT);
    k_bias_add <<<(int)((no + T - 1) / T), T, 0, stream>>>(out, b2, no, D_OUT - 1);
  }
}